// SelfAttentionBlock_70892730188128
// MI455X (gfx1250) — compile-verified
//
#include <hip/hip_runtime.h>
#include <hip/hip_bf16.h>
#include <math.h>

typedef float v2f __attribute__((ext_vector_type(2)));
typedef float v8f __attribute__((ext_vector_type(8)));

#define EMB   1024
#define SEQ   2048
#define NHEAD 16
#define HDIM  64
#define CLM1  1023          // cache rows
#define KVROWS 3071         // CLM1 + SEQ

static __device__ __forceinline__ v8f wmma_f32(v2f a, v2f b, v8f c) {
  // D = A(16x4) * B(4x16) + C, fp32 matrix pipe (V_WMMA_F32_16X16X4_F32)
  return __builtin_amdgcn_wmma_f32_16x16x4_f32(false, a, false, b, (short)0, c,
                                               false, false);
}

// C(M x 1024) = X(M x 1024) @ W(1024 x 1024)^T + bias ; NT layout, both K-contig.
// One wave -> 16(M) x 64(N) tile, 4 accumulators, A reused across 4 Bs.
// Unroll 8 (K=32 per body): 40 b64 loads in flight to hide L2 latency.
__global__ void __launch_bounds__(128) gemm_nt_bias(
    const float* __restrict__ X, const float* __restrict__ W,
    const float* __restrict__ bias, float* __restrict__ C)
{
  const int lane = threadIdx.x & 31;
  const int wave = threadIdx.x >> 5;
  const int half = lane >> 4;       // K sub-split: lanes 0-15 -> K+0/1, 16-31 -> K+2/3
  const int l16  = lane & 15;
  const int m0   = blockIdx.x * 16;
  const int n0   = blockIdx.y * 256 + wave * 64;
  const int koff = 2 * half;

  const float* xr = X + (size_t)(m0 + l16) * EMB + koff;
  const float* w0 = W + (size_t)(n0 +  0 + l16) * EMB + koff;
  const float* w1 = W + (size_t)(n0 + 16 + l16) * EMB + koff;
  const float* w2 = W + (size_t)(n0 + 32 + l16) * EMB + koff;
  const float* w3 = W + (size_t)(n0 + 48 + l16) * EMB + koff;

  v8f a0 = {}, a1 = {}, a2 = {}, a3 = {};
#pragma unroll 8
  for (int k = 0; k < EMB; k += 4) {
    v2f a  = *(const v2f*)(xr + k);
    v2f b0 = *(const v2f*)(w0 + k);
    v2f b1 = *(const v2f*)(w1 + k);
    v2f b2 = *(const v2f*)(w2 + k);
    v2f b3 = *(const v2f*)(w3 + k);
    a0 = wmma_f32(a, b0, a0);
    a1 = wmma_f32(a, b1, a1);
    a2 = wmma_f32(a, b2, a2);
    a3 = wmma_f32(a, b3, a3);
  }
  float bv0 = bias[n0 +  0 + l16];
  float bv1 = bias[n0 + 16 + l16];
  float bv2 = bias[n0 + 32 + l16];
  float bv3 = bias[n0 + 48 + l16];
#pragma unroll
  for (int r = 0; r < 8; ++r) {
    int row = m0 + r + 8 * half;             // C layout: vgpr r -> M = r + 8*half
    float* c = C + (size_t)row * EMB + n0 + l16;
    c[0]  = a0[r] + bv0;
    c[16] = a1[r] + bv1;
    c[32] = a2[r] + bv2;
    c[48] = a3[r] + bv3;
  }
}

// In-place RoPE over a (SEQ x 1024) buffer viewed as (SEQ, 16, 64).
__global__ void __launch_bounds__(256) rope_kernel(float* __restrict__ buf,
                                                   const int* __restrict__ position)
{
  int idx = blockIdx.x * 256 + threadIdx.x;   // SEQ * 512 pair-slots
  if (idx >= SEQ * 512) return;
  int s = idx >> 9;
  int p = idx & 511;
  int n = p >> 5;
  int d = p & 31;
  float pos = (float)(s + position[0]);
  const float kLogBaseOverDim = 0.28782313662425572f;   // ln(10000)/32
  float freq = expf(-(float)d * kLogBaseOverDim);
  float ang  = pos * freq;
  float sn, cs;
  sincosf(ang, &sn, &cs);   // accurate range reduction (angles up to ~2047 rad)
  float* b = buf + (size_t)s * EMB + n * HDIM + 2 * d;
  float xe = b[0], xo = b[1];
  b[0] = xe * cs - xo * sn;
  b[1] = xe * sn + xo * cs;
}

// Flash-style sliding-window attention. One wave per (16-query tile, head).
// Query i attends keys j in [i, i+1023]; 65 key tiles of 16.
__global__ void __launch_bounds__(32) attn_kernel(
    const float* __restrict__ Q,   // 2048 x 1024, rope'd
    const float* __restrict__ Kf,  // 3071 x 1024 (cache + rope'd new k)
    const float* __restrict__ Vf,  // 3071 x 1024
    float* __restrict__ O)         // 2048 x 1024
{
  __shared__ float pshm[256];      // 16x16 P tile for C->A layout transpose
  const int lane  = threadIdx.x & 31;
  const int half  = lane >> 4;
  const int l16   = lane & 15;
  const int i0    = blockIdx.x * 16;
  const int hbase = blockIdx.y * HDIM;

  // Preload Q fragments (16 x 64), reused across all 65 key tiles.
  v2f qa[16];
  const float* qr = Q + (size_t)(i0 + l16) * EMB + hbase + 2 * half;
#pragma unroll
  for (int kk = 0; kk < 16; ++kk) qa[kk] = *(const v2f*)(qr + 4 * kk);

  v8f oacc[4] = {};                // 16 x 64 output accumulator (4 hd tiles)
  float m[8], l[8];
#pragma unroll
  for (int r = 0; r < 8; ++r) { m[r] = -1e30f; l[r] = 0.0f; }

  for (int t = 0; t < 65; ++t) {
    const int j0 = i0 + 16 * t;

    // S = Q K^T for this 16x16 tile (K=64 -> 16 wmma)
    int jr = j0 + l16;
    if (jr > KVROWS - 1) jr = KVROWS - 1;   // clamp; clamped cols are masked
    const float* kr = Kf + (size_t)jr * EMB + hbase + 2 * half;
    v8f s = {};
#pragma unroll
    for (int kk = 0; kk < 16; ++kk) {
      v2f kb = *(const v2f*)(kr + 4 * kk);
      s = wmma_f32(qa[kk], kb, s);
    }

    // Online softmax over the 16 columns (lanes of each half).
    // __expf -> v_exp_f32 (TRANS32, co-executes with the matrix pipe).
    float pv[8], scl[8];
#pragma unroll
    for (int r = 0; r < 8; ++r) {
      int qi = i0 + r + 8 * half;
      int j  = j0 + l16;
      float v = s[r] * 0.125f;     // 1/sqrt(64)
      bool valid = (j >= qi) && (j <= qi + 1023);
      v = valid ? v : -1e30f;
      float mx = v;
      mx = fmaxf(mx, __shfl_xor(mx, 1));
      mx = fmaxf(mx, __shfl_xor(mx, 2));
      mx = fmaxf(mx, __shfl_xor(mx, 4));
      mx = fmaxf(mx, __shfl_xor(mx, 8));
      float mnew = fmaxf(m[r], mx);
      float p = __expf(v - mnew);
      float rs = p;
      rs += __shfl_xor(rs, 1);
      rs += __shfl_xor(rs, 2);
      rs += __shfl_xor(rs, 4);
      rs += __shfl_xor(rs, 8);
      float corr = __expf(m[r] - mnew);
      l[r] = l[r] * corr + rs;
      m[r] = mnew;
      scl[r] = corr;
      pv[r] = p;
    }
    // Rescale O accumulators (score rows and O rows share the same layout).
#pragma unroll
    for (int h4 = 0; h4 < 4; ++h4)
#pragma unroll
      for (int r = 0; r < 8; ++r) oacc[h4][r] *= scl[r];

    // Transpose P (C layout) -> A layout through LDS.
    __syncthreads();
#pragma unroll
    for (int r = 0; r < 8; ++r) pshm[(r + 8 * half) * 16 + l16] = pv[r];
    __syncthreads();

    // O += P(16x16) @ V(16x64): 4 K-steps x 4 hd tiles.
#pragma unroll
    for (int kk = 0; kk < 4; ++kk) {
      v2f pa;
      pa.x = pshm[l16 * 16 + 4 * kk + 2 * half];
      pa.y = pshm[l16 * 16 + 4 * kk + 2 * half + 1];
      int vr0 = j0 + 4 * kk + 2 * half;
      int vr1 = vr0 + 1;
      if (vr0 > KVROWS - 1) vr0 = KVROWS - 1;   // masked cols carry P=0
      if (vr1 > KVROWS - 1) vr1 = KVROWS - 1;
      const float* vb0 = Vf + (size_t)vr0 * EMB + hbase + l16;
      const float* vb1 = Vf + (size_t)vr1 * EMB + hbase + l16;
#pragma unroll
      for (int h4 = 0; h4 < 4; ++h4) {
        v2f b;
        b.x = vb0[h4 * 16];
        b.y = vb1[h4 * 16];
        oacc[h4] = wmma_f32(pa, b, oacc[h4]);
      }
    }
    __syncthreads();
  }

  // Normalize and write o (S, N, H).
#pragma unroll
  for (int r = 0; r < 8; ++r) {
    float inv = 1.0f / l[r];
    int row = i0 + r + 8 * half;
    float* orow = O + (size_t)row * EMB + hbase + l16;
#pragma unroll
    for (int h4 = 0; h4 < 4; ++h4) orow[h4 * 16] = oacc[h4][r] * inv;
  }
}

__global__ void __launch_bounds__(256) copy4_kernel(const float4* __restrict__ src,
                                                    float4* __restrict__ dst, int n4)
{
  int i = blockIdx.x * 256 + threadIdx.x;
  if (i < n4) dst[i] = src[i];
}

extern "C" void kernel_launch(void* const* d_in, const int* in_sizes, int n_in,
                              void* d_out, int out_size, void* d_ws, size_t ws_size,
                              hipStream_t stream)
{
  const float* x  = (const float*)d_in[0];
  const float* Wq = (const float*)d_in[1];
  const float* bq = (const float*)d_in[2];
  const float* Wk = (const float*)d_in[3];
  const float* bk = (const float*)d_in[4];
  const float* Wv = (const float*)d_in[5];
  const float* bv = (const float*)d_in[6];
  const float* Wo = (const float*)d_in[7];
  const float* bo = (const float*)d_in[8];
  const float* kc = (const float*)d_in[9];
  const float* vc = (const float*)d_in[10];
  const int* position = (const int*)d_in[11];
  float* out = (float*)d_out;

  // Workspace layout (fp32): q_buf | kfull | vfull | o_buf  (~42 MB)
  float* q_buf = (float*)d_ws;
  float* kfull = q_buf + (size_t)SEQ * EMB;
  float* vfull = kfull + (size_t)KVROWS * EMB;
  float* o_buf = vfull + (size_t)KVROWS * EMB;

  const int cache4 = (CLM1 * EMB) / 4;
  const int cblocks = (cache4 + 255) / 256;

  // 1) cache -> head of kfull/vfull
  copy4_kernel<<<cblocks, 256, 0, stream>>>((const float4*)kc, (float4*)kfull, cache4);
  copy4_kernel<<<cblocks, 256, 0, stream>>>((const float4*)vc, (float4*)vfull, cache4);

  // 2) Q/K/V projections (k/v land after the cache rows)
  dim3 ggrid(SEQ / 16, EMB / 256);
  gemm_nt_bias<<<ggrid, 128, 0, stream>>>(x, Wq, bq, q_buf);
  gemm_nt_bias<<<ggrid, 128, 0, stream>>>(x, Wk, bk, kfull + (size_t)CLM1 * EMB);
  gemm_nt_bias<<<ggrid, 128, 0, stream>>>(x, Wv, bv, vfull + (size_t)CLM1 * EMB);

  // 3) RoPE on q and the new k rows
  const int rblocks = (SEQ * 512 + 255) / 256;
  rope_kernel<<<rblocks, 256, 0, stream>>>(q_buf, position);
  rope_kernel<<<rblocks, 256, 0, stream>>>(kfull + (size_t)CLM1 * EMB, position);

  // 4) sliding-window flash attention
  attn_kernel<<<dim3(SEQ / 16, NHEAD), 32, 0, stream>>>(q_buf, kfull, vfull, o_buf);

  // 5) output projection -> d_out[0 : SEQ*EMB]
  gemm_nt_bias<<<ggrid, 128, 0, stream>>>(o_buf, Wo, bo, out);

  // 6) new_k / new_v = last 1023 rows of kfull/vfull
  copy4_kernel<<<cblocks, 256, 0, stream>>>((const float4*)(kfull + (size_t)SEQ * EMB),
                                            (float4*)(out + (size_t)SEQ * EMB), cache4);
  copy4_kernel<<<cblocks, 256, 0, stream>>>((const float4*)(vfull + (size_t)SEQ * EMB),
                                            (float4*)(out + (size_t)SEQ * EMB + (size_t)CLM1 * EMB),
                                            cache4);
}